// MGN_58686433132692
// MI455X (gfx1250) — compile-verified
//
#include <hip/hip_runtime.h>

typedef __attribute__((ext_vector_type(2))) float v2f;
typedef __attribute__((ext_vector_type(8))) float v8f;

#define B_ 8
#define N_ 4096
#define D_ 64
#define K_ 16
#define L_ 4
#define PRED_ 64
#define CAT_ ((L_ + 1) * D_) /* 320 */

__device__ __forceinline__ v8f wmma4(v2f a, v2f b, v8f c) {
  return __builtin_amdgcn_wmma_f32_16x16x4_f32(false, a, false, b, (short)0, c,
                                               false, false);
}

// ---------------------------------------------------------------- kNN (d=3)
__global__ void knn3_kernel(const float* __restrict__ coords,
                            int* __restrict__ idx_out) {
  const int b = blockIdx.y;
  const int q = blockIdx.x * blockDim.x + threadIdx.x;
  const float* cb = coords + (size_t)b * N_ * 3;
  const float qx = cb[q * 3 + 0], qy = cb[q * 3 + 1], qz = cb[q * 3 + 2];
  float key[K_]; int val[K_];
#pragma unroll
  for (int i = 0; i < K_; ++i) { key[i] = 3.0e38f; val[i] = 0; }
  __shared__ float sc[256 * 3];
  for (int t0 = 0; t0 < N_; t0 += 256) {
    __syncthreads();
    for (int i = threadIdx.x; i < 256 * 3; i += 256) sc[i] = cb[t0 * 3 + i];
    __syncthreads();
    for (int j = 0; j < 256; ++j) {
      const float dx = qx - sc[j * 3 + 0];
      const float dy = qy - sc[j * 3 + 1];
      const float dz = qz - sc[j * 3 + 2];
      const float d2 = dx * dx + dy * dy + dz * dz;
      const int cj = t0 + j;
      if (cj == q) continue;
      if (d2 < key[K_ - 1]) {
        float dv = d2; int iv = cj;
#pragma unroll
        for (int t = 0; t < K_; ++t) {
          if (dv < key[t]) {
            float tf = key[t]; key[t] = dv; dv = tf;
            int ti = val[t]; val[t] = iv; iv = ti;
          }
        }
      }
    }
  }
  int* op = idx_out + ((size_t)b * N_ + q) * K_;
#pragma unroll
  for (int i = 0; i < K_; ++i) op[i] = val[i];
}

// ------------------------------------------------------- kNN (d=64, WMMA)
// One wave per 32 query rows. Gram tiles via V_WMMA_F32_16X16X4_F32.
__global__ void knn64_kernel(const float* __restrict__ x, int xstride,
                             int* __restrict__ idx_out) {
  const int b = blockIdx.y;
  const int m0 = blockIdx.x * 32;
  const int lane = threadIdx.x;
  const int mh = lane & 15;
  const int h = lane >> 4;
  const float* xb = x + (size_t)b * N_ * xstride;

  __shared__ float qbuf[32 * 64];
  __shared__ float cbuf[16 * 64];
  __shared__ float qn[32];
  __shared__ float cn[16];
  __shared__ float dtile[32][16];

  for (int t = 0; t < 2; ++t) {
    const int row = t * 16 + mh;
    const float* src = xb + (size_t)(m0 + row) * xstride + h * 32;
    float s = 0.f;
#pragma unroll
    for (int p = 0; p < 32; ++p) {
      const float v = src[p];
      qbuf[row * 64 + h * 32 + p] = v;
      s += v * v;
    }
    s += __shfl_xor(s, 16, 32);
    if (h == 0) qn[row] = s;
  }
  __syncthreads();
  const float myqn = qn[lane];

  float key[K_]; int val[K_];
#pragma unroll
  for (int i = 0; i < K_; ++i) { key[i] = 3.0e38f; val[i] = 0; }

  for (int c0 = 0; c0 < N_; c0 += 16) {
    __syncthreads();
    {
      const float* src = xb + (size_t)(c0 + mh) * xstride + h * 32;
      float s = 0.f;
#pragma unroll
      for (int p = 0; p < 32; ++p) {
        const float v = src[p];
        cbuf[mh * 64 + h * 32 + p] = v;
        s += v * v;
      }
      s += __shfl_xor(s, 16, 32);
      if (h == 0) cn[mh] = s;
    }
    __syncthreads();
    v8f acc0 = {};
    v8f acc1 = {};
    const v2f* qp0 = (const v2f*)&qbuf[mh * 64];
    const v2f* qp1 = (const v2f*)&qbuf[(16 + mh) * 64];
    const v2f* cp = (const v2f*)&cbuf[mh * 64];
#pragma unroll
    for (int kk = 0; kk < 16; ++kk) {
      const v2f bv = cp[2 * kk + h];
      acc0 = wmma4(qp0[2 * kk + h], bv, acc0);
      acc1 = wmma4(qp1[2 * kk + h], bv, acc1);
    }
#pragma unroll
    for (int r = 0; r < 8; ++r) {
      const int row = h * 8 + r;
      dtile[row][mh] = acc0[r];
      dtile[16 + row][mh] = acc1[r];
    }
    __syncthreads();
#pragma unroll
    for (int j = 0; j < 16; ++j) {
      const int cj = c0 + j;
      if (cj == m0 + lane) continue;
      const float d2 = myqn + cn[j] - 2.0f * dtile[lane][j];
      if (d2 < key[K_ - 1]) {
        float dv = d2; int iv = cj;
#pragma unroll
        for (int t = 0; t < K_; ++t) {
          if (dv < key[t]) {
            float tf = key[t]; key[t] = dv; dv = tf;
            int ti = val[t]; val[t] = iv; iv = ti;
          }
        }
      }
    }
  }
  int* op = idx_out + ((size_t)b * N_ + m0 + lane) * K_;
#pragma unroll
  for (int i = 0; i < K_; ++i) op[i] = val[i];
}

// ------------------------------------------------------- EdgeConv (WMMA)
// One wave per point: M=16 neighbor rows, two GEMMs, row-max.
template <int KD, bool DCONV, bool ACC>
__global__ void edgeconv_kernel(const float* __restrict__ xin, int xstride,
                                const int* __restrict__ idx,
                                const float* __restrict__ coords,
                                const float* __restrict__ W1,
                                const float* __restrict__ b1,
                                const float* __restrict__ W2,
                                const float* __restrict__ b2,
                                float* __restrict__ xout, int ostride) {
  constexpr int KDR = DCONV ? (2 * D_ + 6) : (2 * D_);
  constexpr int NPAIR = KD / 2;
  __shared__ float w1s[NPAIR * 64 * 2];
  __shared__ float w2s[32 * 64 * 2];
  __shared__ float b1s[64];
  __shared__ float b2s[64];
  __shared__ float ebuf[4][16 * KD];
  __shared__ float h1buf[4][16 * 64];

  for (int i = threadIdx.x; i < NPAIR * 64; i += blockDim.x) {
    const int p = i >> 6, n = i & 63;
    const int k0 = 2 * p, k1 = 2 * p + 1;
    v2f v;
    v.x = (k0 < KDR) ? W1[k0 * 64 + n] : 0.f;
    v.y = (k1 < KDR) ? W1[k1 * 64 + n] : 0.f;
    ((v2f*)w1s)[i] = v;
  }
  for (int i = threadIdx.x; i < 32 * 64; i += blockDim.x) {
    const int p = i >> 6, n = i & 63;
    v2f v;
    v.x = W2[(2 * p) * 64 + n];
    v.y = W2[(2 * p + 1) * 64 + n];
    ((v2f*)w2s)[i] = v;
  }
  if (threadIdx.x < 64) {
    b1s[threadIdx.x] = b1[threadIdx.x];
    b2s[threadIdx.x] = b2[threadIdx.x];
  }
  __syncthreads();

  const int wave = threadIdx.x >> 5;
  const int lane = threadIdx.x & 31;
  const int mh = lane & 15, h = lane >> 4;

  for (int it = 0; it < (B_ * N_) / 2048; ++it) {
    const int pt = it * 2048 + blockIdx.x * 4 + wave;
    const int b = pt >> 12;
    const int p = pt & (N_ - 1);
    const int nidx = idx[(size_t)pt * K_ + mh];
    const float* xrow = xin + ((size_t)b * N_ + p) * xstride;
    const float* nrow = xin + ((size_t)b * N_ + nidx) * xstride;
    float* erow = &ebuf[wave][mh * KD];
    if (h == 0) {
#pragma unroll
      for (int k = 0; k < 64; ++k) erow[k] = xrow[k];
    } else {
#pragma unroll
      for (int k = 0; k < 64; ++k) erow[64 + k] = nrow[k] - xrow[k];
      if (DCONV) {
        const float* crow = coords + ((size_t)b * N_ + p) * 3;
        const float* ncr = coords + ((size_t)b * N_ + nidx) * 3;
        erow[128] = crow[0]; erow[129] = crow[1]; erow[130] = crow[2];
        erow[131] = ncr[0];  erow[132] = ncr[1];  erow[133] = ncr[2];
        erow[134] = 0.f;     erow[135] = 0.f;
      }
    }
    __syncthreads();
    const v2f* ep = (const v2f*)&ebuf[wave][mh * KD];
#pragma unroll
    for (int n0 = 0; n0 < 64; n0 += 16) {
      v8f acc;
      const float bias = b1s[n0 + mh];
#pragma unroll
      for (int r = 0; r < 8; ++r) acc[r] = bias;
      for (int kk = 0; kk < KD / 4; ++kk) {
        const v2f a = ep[2 * kk + h];
        const v2f bv = ((const v2f*)w1s)[(2 * kk + h) * 64 + n0 + mh];
        acc = wmma4(a, bv, acc);
      }
#pragma unroll
      for (int r = 0; r < 8; ++r) {
        float v = acc[r];
        v = v > 0.f ? v : 0.01f * v; // leaky_relu
        h1buf[wave][(h * 8 + r) * 64 + n0 + mh] = v;
      }
    }
    __syncthreads();
    const v2f* hp = (const v2f*)&h1buf[wave][mh * 64];
#pragma unroll
    for (int n0 = 0; n0 < 64; n0 += 16) {
      v8f acc;
      const float bias = b2s[n0 + mh];
#pragma unroll
      for (int r = 0; r < 8; ++r) acc[r] = bias;
#pragma unroll
      for (int kk = 0; kk < 16; ++kk) {
        acc = wmma4(hp[2 * kk + h],
                    ((const v2f*)w2s)[(2 * kk + h) * 64 + n0 + mh], acc);
      }
      float m = acc[0];
#pragma unroll
      for (int r = 1; r < 8; ++r) m = fmaxf(m, acc[r]);
      m = fmaxf(m, __shfl_xor(m, 16, 32));
      if (h == 0) {
        float* o = xout + ((size_t)b * N_ + p) * ostride + n0 + mh;
        if (ACC) *o += m; else *o = m;
      }
    }
    __syncthreads();
  }
}

// ------------------------------------------------------------- tail kernels
__global__ void copy_feats_kernel(const float* __restrict__ feats,
                                  float* __restrict__ xcat) {
  const size_t i = (size_t)blockIdx.x * 256 + threadIdx.x;
  const size_t pt = i >> 6;
  const int k = (int)(i & 63);
  xcat[pt * CAT_ + k] = feats[i];
}

__global__ void site_kernel(const float* __restrict__ xcat,
                            const float* __restrict__ Ws,
                            const float* __restrict__ bs,
                            float* __restrict__ xsite) {
  __shared__ float row[CAT_];
  const int pt = blockIdx.x;
  const float* src = xcat + (size_t)pt * CAT_;
  for (int i = threadIdx.x; i < CAT_; i += 64) row[i] = src[i];
  __syncthreads();
  const int n = threadIdx.x;
  float s = bs[n];
  for (int k = 0; k < CAT_; ++k) s += row[k] * Ws[k * PRED_ + n];
  xsite[(size_t)pt * PRED_ + n] = s;
}

__global__ void pool_kernel(const float* __restrict__ xsite,
                            float* __restrict__ xpool) {
  const int b = blockIdx.x >> 6;
  const int n = blockIdx.x & 63;
  float m = -3.0e38f;
  for (int i = threadIdx.x; i < N_; i += 256)
    m = fmaxf(m, xsite[((size_t)b * N_ + i) * PRED_ + n]);
  __shared__ float red[256];
  red[threadIdx.x] = m;
  __syncthreads();
  for (int s = 128; s > 0; s >>= 1) {
    if (threadIdx.x < s)
      red[threadIdx.x] = fmaxf(red[threadIdx.x], red[threadIdx.x + s]);
    __syncthreads();
  }
  if (threadIdx.x == 0) xpool[blockIdx.x] = red[0];
}

__global__ void head_kernel(const float* __restrict__ xpool,
                            const float* __restrict__ Wc1,
                            const float* __restrict__ bc1,
                            const float* __restrict__ Wc2,
                            const float* __restrict__ bc2,
                            float* __restrict__ out) {
  __shared__ float hbuf[B_][PRED_];
  const int t = threadIdx.x;
  for (int b = 0; b < B_; ++b) {
    float s = bc1[t];
    for (int k = 0; k < PRED_; ++k) s += xpool[b * PRED_ + k] * Wc1[k * PRED_ + t];
    hbuf[b][t] = s > 0.f ? s : 0.01f * s;
  }
  __syncthreads();
  if (t < B_ * 3) {
    const int b = t / 3, c = t % 3;
    float s = bc2[c];
    for (int j = 0; j < PRED_; ++j) s += hbuf[b][j] * Wc2[j * 3 + c];
    out[b * 3 + c] = s;
  }
}

// ------------------------------------------------------------------ driver
extern "C" void kernel_launch(void* const* d_in, const int* in_sizes, int n_in,
                              void* d_out, int out_size, void* d_ws,
                              size_t ws_size, hipStream_t stream) {
  const float* feats  = (const float*)d_in[0];
  const float* coords = (const float*)d_in[1];
  const float* Wd1 = (const float*)d_in[2];
  const float* bd1 = (const float*)d_in[3];
  const float* Wd2 = (const float*)d_in[4];
  const float* bd2 = (const float*)d_in[5];
  const float* Wf1 = (const float*)d_in[6];
  const float* bf1 = (const float*)d_in[7];
  const float* Wf2 = (const float*)d_in[8];
  const float* bf2 = (const float*)d_in[9];
  const float* Ws  = (const float*)d_in[10];
  const float* bs  = (const float*)d_in[11];
  const float* Wc1 = (const float*)d_in[12];
  const float* bc1 = (const float*)d_in[13];
  const float* Wc2 = (const float*)d_in[14];
  const float* bc2 = (const float*)d_in[15];
  float* out = (float*)d_out;

  char* ws = (char*)d_ws;
  int* idx_dist = (int*)ws;   ws += (size_t)B_ * N_ * K_ * 4;
  int* idx_feat = (int*)ws;   ws += (size_t)B_ * N_ * K_ * 4;
  float* xcat   = (float*)ws; ws += (size_t)B_ * N_ * CAT_ * 4;
  float* xsite  = (float*)ws; ws += (size_t)B_ * N_ * PRED_ * 4;
  float* xpool  = (float*)ws;

  knn3_kernel<<<dim3(N_ / 256, B_), 256, 0, stream>>>(coords, idx_dist);
  copy_feats_kernel<<<(B_ * N_ * D_) / 256, 256, 0, stream>>>(feats, xcat);

  for (int l = 0; l < L_; ++l) {
    const float* xin = xcat + l * D_;
    float* xo = xcat + (l + 1) * D_;
    knn64_kernel<<<dim3(N_ / 32, B_), 32, 0, stream>>>(xin, CAT_, idx_feat);
    edgeconv_kernel<128, false, false><<<512, 128, 0, stream>>>(
        xin, CAT_, idx_feat, nullptr,
        Wf1 + (size_t)l * 128 * 64, bf1 + l * 64,
        Wf2 + (size_t)l * 64 * 64, bf2 + l * 64, xo, CAT_);
    edgeconv_kernel<136, true, true><<<512, 128, 0, stream>>>(
        xin, CAT_, idx_dist, coords,
        Wd1 + (size_t)l * 134 * 64, bd1 + l * 64,
        Wd2 + (size_t)l * 64 * 64, bd2 + l * 64, xo, CAT_);
  }

  site_kernel<<<B_ * N_, 64, 0, stream>>>(xcat, Ws, bs, xsite);
  pool_kernel<<<B_ * PRED_, 256, 0, stream>>>(xsite, xpool);
  head_kernel<<<1, 64, 0, stream>>>(xpool, Wc1, bc1, Wc2, bc2, out);
}